// LoMoEOutputHead_10642928959990
// MI455X (gfx1250) — compile-verified
//
#include <hip/hip_runtime.h>
#include <hip/hip_bf16.h>

typedef __attribute__((ext_vector_type(2))) float v2f;
typedef __attribute__((ext_vector_type(8))) float v8f;

#define B_   32
#define V_   21
#define D_   512
#define P_   64
#define IN_  (D_ * P_)      // 32768
#define OUT_ 96
#define E_   8
#define R_   8
#define H_   256
#define MROWS (B_ * V_)     // 672
#define NCOMB (OUT_ + E_ * R_) // 160
#define MTILES (MROWS / 16)    // 42
#define MPAIRS (MTILES / 2)    // 21 (two M-tiles per wave)
#define NTILES (NCOMB / 16)    // 10
#define KSPLIT 32
#define KCHUNK (IN_ / KSPLIT)  // 1024
#define SCALING 2.0f           // 16/8

// ---------------- workspace layout (floats) ----------------
// acc:    [672][160]   at 0            (107520)
// pooled: [32][512]    at 107520       (16384)
// wfull:  [32][8]      at 123904       (256)
#define WS_ACC    0
#define WS_POOLED 107520
#define WS_WFULL  (107520 + 16384)
#define WS_TOTAL  (107520 + 16384 + 256)

__global__ void zero_ws_kernel(float* __restrict__ ws, int n) {
    int i = blockIdx.x * blockDim.x + threadIdx.x;
    if (i < n) ws[i] = 0.0f;
}

// pooled[b][d] = mean over (v,p) of x[b][v][d][p] ; one wave per (b,d)
__global__ void pool_kernel(const float* __restrict__ x, float* __restrict__ pooled) {
    int wid  = (blockIdx.x * blockDim.x + threadIdx.x) >> 5;
    int lane = threadIdx.x & 31;
    if (wid >= B_ * D_) return;
    int b = wid >> 9;          // /512
    int d = wid & (D_ - 1);
    const float* base = x + ((size_t)b * V_ * D_ + d) * P_;
    float s = 0.0f;
    for (int v = 0; v < V_; ++v) {
        const float* p = base + (size_t)v * D_ * P_;
        s += p[lane] + p[lane + 32];
    }
    #pragma unroll
    for (int off = 16; off > 0; off >>= 1) s += __shfl_down(s, off, 32);
    if (lane == 0) pooled[wid] = s * (1.0f / (float)(V_ * P_));
}

// Router MLP + softmax + top-2 -> wfull[B][E], probs -> out tail. One block of 256.
__global__ void router_kernel(const float* __restrict__ pooled,
                              const float* __restrict__ W1, const float* __restrict__ b1,
                              const float* __restrict__ W2, const float* __restrict__ b2,
                              float* __restrict__ wfull, float* __restrict__ probs_out) {
    __shared__ float sh_h[B_ * H_];       // 32 KB
    __shared__ float sh_logit[B_ * E_];
    int t = threadIdx.x;                  // 0..255
    // h[b][t] = relu(pooled[b] . W1[t] + b1[t]) for all b
    for (int b = 0; b < B_; ++b) {
        float acc = b1[t];
        const float* pr = pooled + (size_t)b * D_;
        const float* wr = W1 + (size_t)t * D_;
        for (int k = 0; k < D_; ++k) acc += pr[k] * wr[k];
        sh_h[b * H_ + t] = acc > 0.0f ? acc : 0.0f;
    }
    __syncthreads();
    // logits[b][e] ; 256 threads cover 32*8
    {
        int b = t >> 3, e = t & 7;
        float acc = b2[e];
        const float* hr = sh_h + b * H_;
        const float* wr = W2 + (size_t)e * H_;
        for (int k = 0; k < H_; ++k) acc += hr[k] * wr[k];
        sh_logit[b * E_ + e] = acc;
    }
    __syncthreads();
    if (t < B_) {
        int b = t;
        float lg[E_], mx = -1e30f;
        for (int e = 0; e < E_; ++e) { lg[e] = sh_logit[b * E_ + e]; mx = fmaxf(mx, lg[e]); }
        float sum = 0.0f;
        for (int e = 0; e < E_; ++e) { lg[e] = __expf(lg[e] - mx); sum += lg[e]; }
        float inv = 1.0f / sum;
        float pr[E_];
        for (int e = 0; e < E_; ++e) { pr[e] = lg[e] * inv; probs_out[b * E_ + e] = pr[e]; }
        // top-2
        int i1 = 0; for (int e = 1; e < E_; ++e) if (pr[e] > pr[i1]) i1 = e;
        int i2 = (i1 == 0) ? 1 : 0;
        for (int e = 0; e < E_; ++e) if (e != i1 && pr[e] > pr[i2]) i2 = e;
        float w1 = pr[i1], w2 = pr[i2];
        float ws = w1 + w2; ws = ws < 1e-6f ? 1e-6f : ws;
        float* wf = wfull + b * E_;
        wf[i1] = w1 / ws;
        wf[i2] = w2 / ws;
    }
}

// Fused split-K WMMA GEMM: acc[672][160] += flat[672][32768] x Wcomb[32768][160]
// Wcomb column n: n<96 -> W_base[n][:], else lora_A[(n-96)][:]  (both row-major [n][k])
// One wave per (mtile-PAIR, ntile, kchunk): B fragment loaded once, used by two
// V_WMMA_F32_16X16X4_F32 (two M-tiles) -> 3 loads per 2 WMMAs instead of 4.
__global__ __launch_bounds__(32)
void gemm_wmma_kernel(const float* __restrict__ flat,
                      const float* __restrict__ Wb,
                      const float* __restrict__ lA,
                      float* __restrict__ acc) {
    int bid = blockIdx.x;
    int kc = bid % KSPLIT; bid /= KSPLIT;
    int nt = bid % NTILES; bid /= NTILES;
    int mp = bid;                 // 0..20  (M-tile pair)
    int lane = threadIdx.x;       // 0..31
    int m    = lane & 15;
    int khi  = lane >> 4;         // K sub-pair select (0: K=0,1 ; 1: K=2,3)

    // A fragments: lane(16*khi+m) holds flat[mt*16+m][k + 2*khi + {0,1}]
    const float* arow0 = flat + (size_t)((2 * mp + 0) * 16 + m) * IN_;
    const float* arow1 = flat + (size_t)((2 * mp + 1) * 16 + m) * IN_;
    // B fragment: lane(16*khi+n) holds Wcomb_col(nt*16+n)[k + 2*khi + {0,1}]
    int ncol = nt * 16 + m;
    const float* brow = (ncol < OUT_) ? (Wb + (size_t)ncol * IN_)
                                      : (lA + (size_t)(ncol - OUT_) * IN_);
    int kbase = kc * KCHUNK + 2 * khi;

    v8f c0 = {0.f, 0.f, 0.f, 0.f, 0.f, 0.f, 0.f, 0.f};
    v8f c1 = {0.f, 0.f, 0.f, 0.f, 0.f, 0.f, 0.f, 0.f};
    #pragma unroll 4
    for (int k = 0; k < KCHUNK; k += 4) {
        v2f b  = *(const v2f*)(brow  + kbase + k);
        v2f a0 = *(const v2f*)(arow0 + kbase + k);
        v2f a1 = *(const v2f*)(arow1 + kbase + k);
        c0 = __builtin_amdgcn_wmma_f32_16x16x4_f32(
                false, a0, false, b, (short)0, c0, false, false);
        c1 = __builtin_amdgcn_wmma_f32_16x16x4_f32(
                false, a1, false, b, (short)0, c1, false, false);
    }

    // C/D layout: VGPR r -> row M = r + 8*khi, col N = lane&15
    float* ap0 = acc + (size_t)((2 * mp + 0) * 16 + 8 * khi) * NCOMB + nt * 16 + (lane & 15);
    float* ap1 = acc + (size_t)((2 * mp + 1) * 16 + 8 * khi) * NCOMB + nt * 16 + (lane & 15);
    #pragma unroll
    for (int r = 0; r < 8; ++r)
        atomicAdd(ap0 + (size_t)r * NCOMB, c0[r]);
    #pragma unroll
    for (int r = 0; r < 8; ++r)
        atomicAdd(ap1 + (size_t)r * NCOMB, c1[r]);
}

// out[row][o] = acc_base + b_base[o] + SCALING * sum_e wfull[b][e] * (lora_B[e][o][:] . acc_temp[row][e][:])
__global__ void finalize_kernel(const float* __restrict__ acc,
                                const float* __restrict__ wfull,
                                const float* __restrict__ lora_B,
                                const float* __restrict__ b_base,
                                float* __restrict__ out) {
    int idx = blockIdx.x * blockDim.x + threadIdx.x;
    if (idx >= MROWS * OUT_) return;
    int o = idx % OUT_, row = idx / OUT_;
    int b = row / V_;
    const float* arow = acc + (size_t)row * NCOMB;
    float r = arow[o] + b_base[o];
    #pragma unroll
    for (int e = 0; e < E_; ++e) {
        float w = wfull[b * E_ + e];
        const float* lb = lora_B + ((size_t)e * OUT_ + o) * R_;
        const float* tr = arow + OUT_ + e * R_;
        float d = 0.0f;
        #pragma unroll
        for (int rr = 0; rr < R_; ++rr) d += lb[rr] * tr[rr];
        r += w * SCALING * d;
    }
    out[idx] = r;
}

extern "C" void kernel_launch(void* const* d_in, const int* in_sizes, int n_in,
                              void* d_out, int out_size, void* d_ws, size_t ws_size,
                              hipStream_t stream) {
    const float* x      = (const float*)d_in[0];
    const float* W_base = (const float*)d_in[1];
    const float* b_base = (const float*)d_in[2];
    const float* W1     = (const float*)d_in[3];
    const float* b1     = (const float*)d_in[4];
    const float* W2     = (const float*)d_in[5];
    const float* b2     = (const float*)d_in[6];
    const float* lora_A = (const float*)d_in[7];
    const float* lora_B = (const float*)d_in[8];
    float* out = (float*)d_out;
    float* ws  = (float*)d_ws;

    float* acc    = ws + WS_ACC;
    float* pooled = ws + WS_POOLED;
    float* wfull  = ws + WS_WFULL;

    // 1) zero accumulator + router scratch (harness poisons d_ws)
    zero_ws_kernel<<<(WS_TOTAL + 255) / 256, 256, 0, stream>>>(ws, WS_TOTAL);

    // 2) pool x -> pooled[32][512]  (one wave per (b,d); 16384 waves)
    pool_kernel<<<(B_ * D_ * 32 + 255) / 256, 256, 0, stream>>>(x, pooled);

    // 3) router MLP/softmax/top2; probs written to out tail [672*96 ..)
    router_kernel<<<1, 256, 0, stream>>>(pooled, W1, b1, W2, b2, wfull,
                                         out + MROWS * OUT_);

    // 4) fused base+LoRA GEMM with split-K WMMA, atomics into acc
    gemm_wmma_kernel<<<MPAIRS * NTILES * KSPLIT, 32, 0, stream>>>(x, W_base, lora_A, acc);

    // 5) combine -> out[672][96]
    finalize_kernel<<<(MROWS * OUT_ + 255) / 256, 256, 0, stream>>>(acc, wfull, lora_B,
                                                                    b_base, out);
}